// SVDKMuyGPs_4526895530617
// MI455X (gfx1250) — compile-verified
//
#include <hip/hip_runtime.h>
#include <hip/hip_bf16.h>
#include <math.h>

typedef __attribute__((ext_vector_type(2))) float v2f;
typedef __attribute__((ext_vector_type(8))) float v8f;

#define CN      100000   // N points
#define CDIN    40
#define CDH     30
#define CDOUT   10
#define CB      10000    // batches
#define CNN     64       // neighbors
#define CR      10       // targets
#define CEPS    1e-5f
#define KPAD    12       // K=10 padded to 12 for 3 x (16x16x4) WMMA steps

// ---------------------------------------------------------------------------
// Kernel 1: MLP embedding  emb = prelu(prelu(x W1^T + b1, a1) W2^T + b2, a2)
// Weights staged in LDS; one thread per input row.
// ---------------------------------------------------------------------------
__global__ __launch_bounds__(256) void mlp_kernel(
    const float* __restrict__ x,
    const float* __restrict__ W1, const float* __restrict__ b1,
    const float* __restrict__ a1p,
    const float* __restrict__ W2, const float* __restrict__ b2,
    const float* __restrict__ a2p,
    float* __restrict__ emb, int n)
{
    __shared__ float sW1[CDH * CDIN];
    __shared__ float sW2[CDOUT * CDH];
    __shared__ float sb1[CDH];
    __shared__ float sb2[CDOUT];

    for (int i = threadIdx.x; i < CDH * CDIN; i += 256) sW1[i] = W1[i];
    for (int i = threadIdx.x; i < CDOUT * CDH; i += 256) sW2[i] = W2[i];
    if (threadIdx.x < CDH)   sb1[threadIdx.x] = b1[threadIdx.x];
    if (threadIdx.x < CDOUT) sb2[threadIdx.x] = b2[threadIdx.x];
    __syncthreads();

    int i = blockIdx.x * 256 + threadIdx.x;
    if (i >= n) return;

    const float a1 = a1p[0];
    const float a2 = a2p[0];

    float xi[CDIN];
#pragma unroll
    for (int c = 0; c < CDIN; ++c) xi[c] = x[i * CDIN + c];

    float h[CDH];
#pragma unroll 2
    for (int j = 0; j < CDH; ++j) {
        float acc = sb1[j];
#pragma unroll
        for (int c = 0; c < CDIN; ++c) acc = fmaf(sW1[j * CDIN + c], xi[c], acc);
        h[j] = (acc > 0.0f) ? acc : a1 * acc;
    }
#pragma unroll
    for (int r = 0; r < CDOUT; ++r) {
        float acc = sb2[r];
#pragma unroll
        for (int j = 0; j < CDH; ++j) acc = fmaf(sW2[r * CDH + j], h[j], acc);
        emb[i * CDOUT + r] = (acc > 0.0f) ? acc : a2 * acc;
    }
}

// ---------------------------------------------------------------------------
// Kernel 2: per-batch Gram (fp32 WMMA) -> Matern-1/2 kernel -> Cholesky solve
// One workgroup (128 threads = 4 wave32) per batch.
// ---------------------------------------------------------------------------
__global__ __launch_bounds__(128) void batch_kernel(
    const float* __restrict__ emb,
    const int*   __restrict__ bidx,
    const int*   __restrict__ bnn,
    const float* __restrict__ nnt,      // (B, NN, R)
    float* __restrict__ predictions,    // (B, R)
    float* __restrict__ variances,      // (B)
    float* __restrict__ sigma_partial)  // (B, R)
{
    __shared__ float sXnn[CNN][KPAD];   // 64 x 12 (cols 10,11 zero)
    __shared__ float sK[CNN][CNN + 1];  // stride 65 -> conflict-free rows & cols
    __shared__ float sSol[CNN][12];     // 11 used: R targets + Kcross column
    __shared__ float sXb[16];
    __shared__ float sKc[CNN];
    __shared__ float sSq[CNN];
    __shared__ float sInvD;

    const int b    = blockIdx.x;
    const int tid  = threadIdx.x;
    const int wv   = tid >> 5;          // wave id 0..3
    const int lane = tid & 31;
    const int half = lane >> 4;         // 0: lanes 0-15, 1: lanes 16-31
    const int ml   = lane & 15;

    // --- load neighbor embeddings (zero-padded in K) and the center point ---
    for (int idx = tid; idx < CNN * KPAD; idx += 128) {
        int row = idx / KPAD, c = idx % KPAD;
        float v = 0.0f;
        if (c < CDOUT) v = emb[bnn[b * CNN + row] * CDOUT + c];
        sXnn[row][c] = v;
    }
    if (tid < 16) {
        sXb[tid] = (tid < CDOUT) ? emb[bidx[b] * CDOUT + tid] : 0.0f;
    }
    __syncthreads();

    // --- squared norms ---
    if (tid < CNN) {
        float s = 0.0f;
#pragma unroll
        for (int c = 0; c < CDOUT; ++c) s = fmaf(sXnn[tid][c], sXnn[tid][c], s);
        sSq[tid] = s;
    }
    __syncthreads();

    // --- Gram = Xnn * Xnn^T via V_WMMA_F32_16X16X4_F32, then kernel matrix ---
    // Wave wv owns row-block wv (rows 16wv..16wv+15), iterates column blocks.
    const int rowA = (wv << 4) + ml;
#pragma unroll
    for (int jb = 0; jb < 4; ++jb) {
        const int rowB = (jb << 4) + ml;
        v8f acc = {};
#pragma unroll
        for (int kc = 0; kc < 3; ++kc) {
            const int k0 = kc * 4 + 2 * half;   // A/B 16x4 f32 lane/VGPR layout
            v2f av, bv;
            av.x = sXnn[rowA][k0];
            av.y = sXnn[rowA][k0 + 1];
            bv.x = sXnn[rowB][k0];
            bv.y = sXnn[rowB][k0 + 1];
            acc = __builtin_amdgcn_wmma_f32_16x16x4_f32(
                false, av, false, bv, (short)0, acc, false, false);
        }
        // C/D layout: VGPR v -> M = v + 8*half ; N = lane & 15
#pragma unroll
        for (int v = 0; v < 8; ++v) {
            const int row = (wv << 4) + v + 8 * half;
            const int col = (jb << 4) + ml;
            float g  = acc[v];
            float d2 = sSq[row] + sSq[col] - 2.0f * g;
            float d  = sqrtf(fmaxf(d2, 0.0f));
            float kv = __expf(-d);
            if (row == col) kv += CEPS;
            sK[row][col] = kv;
        }
    }

    // --- cross kernel Kcross[n] = exp(-||xb - xnn[n]||) ---
    if (tid < CNN) {
        float sqb = 0.0f, dot = 0.0f;
#pragma unroll
        for (int c = 0; c < CDOUT; ++c) {
            sqb = fmaf(sXb[c], sXb[c], sqb);
            dot = fmaf(sXb[c], sXnn[tid][c], dot);
        }
        float d2 = sqb + sSq[tid] - 2.0f * dot;
        sKc[tid] = __expf(-sqrtf(fmaxf(d2, 0.0f)));
    }
    __syncthreads();

    // --- rhs = [targets | Kcross] ---
    for (int idx = tid; idx < CNN * (CR + 1); idx += 128) {
        int row = idx / (CR + 1), c = idx % (CR + 1);
        sSol[row][c] = (c < CR) ? nnt[(b * CNN + row) * CR + c] : sKc[row];
    }
    __syncthreads();

    // --- in-place Cholesky of sK (lower triangle), thread = row ---
    for (int j = 0; j < CNN; ++j) {
        if (tid == 0) {
            float dd = sqrtf(sK[j][j]);
            sK[j][j] = dd;
            sInvD = 1.0f / dd;
        }
        __syncthreads();
        if (tid < CNN && tid > j) sK[tid][j] *= sInvD;
        __syncthreads();
        if (tid < CNN && tid > j) {
            float lij = sK[tid][j];
            for (int k = j + 1; k <= tid; ++k)
                sK[tid][k] = fmaf(-lij, sK[k][j], sK[tid][k]);
        }
        __syncthreads();
    }

    // --- forward substitution: L y = rhs ---
    for (int j = 0; j < CNN; ++j) {
        if (tid < CR + 1) sSol[j][tid] /= sK[j][j];
        __syncthreads();
        if (tid < CNN && tid > j) {
            float lij = sK[tid][j];
#pragma unroll
            for (int r = 0; r < CR + 1; ++r)
                sSol[tid][r] = fmaf(-lij, sSol[j][r], sSol[tid][r]);
        }
        __syncthreads();
    }

    // --- backward substitution: L^T x = y ---
    for (int j = CNN - 1; j >= 0; --j) {
        if (tid < CR + 1) sSol[j][tid] /= sK[j][j];
        __syncthreads();
        if (tid < j) {
            float lji = sK[j][tid];   // L^T[tid][j]
#pragma unroll
            for (int r = 0; r < CR + 1; ++r)
                sSol[tid][r] = fmaf(-lji, sSol[j][r], sSol[tid][r]);
        }
        __syncthreads();
    }

    // --- outputs ---
    if (tid < CR) {
        float p = 0.0f, sp = 0.0f;
        for (int n = 0; n < CNN; ++n) {
            p  = fmaf(sKc[n], sSol[n][tid], p);
            sp = fmaf(nnt[(b * CNN + n) * CR + tid], sSol[n][tid], sp);
        }
        predictions[b * CR + tid]   = p;
        sigma_partial[b * CR + tid] = sp;
    }
    if (tid == 32) {
        float v = 0.0f;
        for (int n = 0; n < CNN; ++n) v = fmaf(sKc[n], sSol[n][CR], v);
        variances[b] = 1.0f - v;
    }
}

// ---------------------------------------------------------------------------
// Kernel 3: deterministic reduction of sigma partials (one block per r)
// ---------------------------------------------------------------------------
__global__ __launch_bounds__(256) void sigma_reduce(
    const float* __restrict__ partial, float* __restrict__ out)
{
    __shared__ float red[256];
    const int r = blockIdx.x;
    float s = 0.0f;
    for (int i = threadIdx.x; i < CB; i += 256) s += partial[i * CR + r];
    red[threadIdx.x] = s;
    __syncthreads();
    for (int w = 128; w > 0; w >>= 1) {
        if (threadIdx.x < w) red[threadIdx.x] += red[threadIdx.x + w];
        __syncthreads();
    }
    if (threadIdx.x == 0)
        out[r] = red[0] / (float)(CB * CNN);
}

// ---------------------------------------------------------------------------
extern "C" void kernel_launch(void* const* d_in, const int* in_sizes, int n_in,
                              void* d_out, int out_size, void* d_ws, size_t ws_size,
                              hipStream_t stream)
{
    const float* x    = (const float*)d_in[0];
    const int*   bidx = (const int*)  d_in[1];
    const int*   bnn  = (const int*)  d_in[2];
    // d_in[3] = batch_targets (unused by the reference computation)
    const float* nnt  = (const float*)d_in[4];
    const float* W1   = (const float*)d_in[5];
    const float* b1   = (const float*)d_in[6];
    const float* a1   = (const float*)d_in[7];
    const float* W2   = (const float*)d_in[8];
    const float* b2   = (const float*)d_in[9];
    const float* a2   = (const float*)d_in[10];

    float* out          = (float*)d_out;
    float* predictions  = out;                  // B*R
    float* variances    = out + CB * CR;        // B
    float* sigma        = out + CB * CR + CB;   // R

    float* emb           = (float*)d_ws;        // N*DOUT floats
    float* sigma_partial = emb + (size_t)CN * CDOUT;  // B*R floats

    mlp_kernel<<<(CN + 255) / 256, 256, 0, stream>>>(
        x, W1, b1, a1, W2, b2, a2, emb, CN);
    batch_kernel<<<CB, 128, 0, stream>>>(
        emb, bidx, bnn, nnt, predictions, variances, sigma_partial);
    sigma_reduce<<<CR, 256, 0, stream>>>(sigma_partial, sigma);
}